// ThalamicRNN_18116172054562
// MI455X (gfx1250) — compile-verified
//
#include <hip/hip_runtime.h>
#include <hip/hip_bf16.h>
#include <cmath>

// ---------------------------------------------------------------------------
// ThalamicRNN step on MI455X (gfx1250), fp32 precision via V_WMMA_F32_16X16X4_F32
//   ext  = stim @ I_stim                      (2048x128 @ 128x2048)
//   rvs  = r_th * (r @ V^T)                   (2048x64)
//   pre  = -x + r@J + rvs@U^T + B + ext
//   x'   = x + (DT/TAU)*(pre + SIGMA*eps)
//   r'   = softplus(x')
//
// rnn_main is register-blocked 32x64 per wave (2 A-frags x 4 B-frags, 8 C
// accumulators) to cut per-WMMA load traffic from 16 B/lane to 6 B/lane.
// ---------------------------------------------------------------------------

typedef __attribute__((ext_vector_type(2))) float v2f;
typedef __attribute__((ext_vector_type(8))) float v8f;

#define NNEUR 2048
#define RANKD 64
#define BATCHD 2048
#define INDIM 128

#define DT_TAU 0.33333333333333f
#define SIGMA_C 0.816496580927726f  // sqrt(2*0.05/0.15)

#define WMMA_F32(a, b, c) \
    __builtin_amdgcn_wmma_f32_16x16x4_f32(false, (a), false, (b), (short)0, (c), false, false)

// A-fragment (16x4, MxK) / k-major-stored B: lane holds row (lane&15), two
// consecutive K values at k + 2*(lane>=16).  (ISA 7.12.2)
__device__ __forceinline__ v2f frag_kmajor(const float* __restrict__ p,
                                           int row_base, int ld, int k, int lane) {
    const float* q = p + (size_t)(row_base + (lane & 15)) * (size_t)ld
                       + (size_t)(k + ((lane >> 4) << 1));
    return v2f{q[0], q[1]};
}

// ---------------------------------------------------------------------------
// Kernel 1: rvs[b,j] = r_th[b,j] * sum_k r[b,k] * V[j,k]
//   GEMM M=2048, N=64, K=2048.  B[k,j] = V[j*2048+k] -> k-major per lane.
//   512 tiles of 16x16, one per wave; 64 blocks x 8 waves.
// ---------------------------------------------------------------------------
__global__ __launch_bounds__(256) void rnn_rv_kernel(
        const float* __restrict__ r,      // (2048, 2048)
        const float* __restrict__ V,      // (64, 2048)
        const float* __restrict__ rth,    // (2048, 64)
        float* __restrict__ rvs) {        // (2048, 64)
    const int lane = threadIdx.x & 31;
    const int wave = threadIdx.x >> 5;
    const int tile = blockIdx.x * 8 + wave;       // 0..511
    const int Mb = (tile >> 2) << 4;              // batch tile
    const int Jb = (tile & 3) << 4;               // rank tile

    v8f c = {};
#pragma unroll 8
    for (int k = 0; k < NNEUR; k += 4) {
        v2f a = frag_kmajor(r, Mb, NNEUR, k, lane);
        v2f b = frag_kmajor(V, Jb, NNEUR, k, lane);   // V^T fragment
        c = WMMA_F32(a, b, c);
    }

    const int j  = Jb + (lane & 15);
    const int m0 = Mb + ((lane >> 4) << 3);       // lanes 16..31 -> M+8
#pragma unroll
    for (int i = 0; i < 8; ++i) {
        const size_t idx = (size_t)(m0 + i) * RANKD + j;
        rvs[idx] = c[i] * rth[idx];
    }
}

// ---------------------------------------------------------------------------
// Kernel 2: fused main step, register-blocked 32x64 per wave.
//   c[2][4] += r@J (K=2048) + stim@I_stim (K=128) + rvs@U^T (K=64)
//   2048 wave-tiles (64 M-tiles x 32 N-tiles) -> 256 blocks x 8 waves.
// ---------------------------------------------------------------------------
__global__ __launch_bounds__(256) void rnn_main_kernel(
        const float* __restrict__ r,      // (2048, 2048)
        const float* __restrict__ J,      // (2048, 2048)
        const float* __restrict__ stim,   // (2048, 128)
        const float* __restrict__ Ist,    // (128, 2048)
        const float* __restrict__ rvs,    // (2048, 64)   from kernel 1
        const float* __restrict__ U,      // (2048, 64)   B[j,n] = U[n*64+j]
        const float* __restrict__ x,      // (2048, 2048)
        const float* __restrict__ Bb,     // (1, 2048)
        const float* __restrict__ eps,    // (2048, 2048)
        float* __restrict__ out) {        // (2*2048*2048): x_new then r_new
    const int lane = threadIdx.x & 31;
    const int wave = threadIdx.x >> 5;
    const int tile = blockIdx.x * 8 + wave;       // 0..2047
    const int Mb = (tile >> 5) << 5;              // 64 M-tiles of 32 rows
    const int Nb = (tile & 31) << 6;              // 32 N-tiles of 64 cols

    const int lr = lane & 15;                     // row/col within fragment
    const int lk = (lane >> 4) << 1;              // K sub-offset (0 or 2)

    v8f c[2][4] = {};

    // ---- r @ J : dominant GEMM, K = 2048 -------------------------------
    {
        const float* a0 = r + (size_t)(Mb + lr)      * NNEUR + lk;   // M rows 0..15
        const float* a1 = r + (size_t)(Mb + 16 + lr) * NNEUR + lk;   // M rows 16..31
        const float* bp = J + (size_t)lk * NNEUR + (Nb + lr);        // rows lk, cols
#pragma unroll 2
        for (int k = 0; k < NNEUR; k += 4) {
            // stream-ahead prefetch (gfx1250 global_prefetch_b8)
            __builtin_prefetch(a0 + k + 128, 0, 1);
            __builtin_prefetch(bp + (size_t)(k + 32) * NNEUR, 0, 1);

            v2f fa0 = *(const v2f*)(a0 + k);
            v2f fa1 = *(const v2f*)(a1 + k);
            v2f fb[4];
#pragma unroll
            for (int j = 0; j < 4; ++j) {
                const float* q = bp + (size_t)k * NNEUR + j * 16;
                fb[j] = v2f{q[0], q[NNEUR]};
            }
#pragma unroll
            for (int j = 0; j < 4; ++j) {
                c[0][j] = WMMA_F32(fa0, fb[j], c[0][j]);
                c[1][j] = WMMA_F32(fa1, fb[j], c[1][j]);
            }
        }
    }

    // ---- stim @ I_stim : K = 128 ---------------------------------------
    {
        const float* a0 = stim + (size_t)(Mb + lr)      * INDIM + lk;
        const float* a1 = stim + (size_t)(Mb + 16 + lr) * INDIM + lk;
        const float* bp = Ist + (size_t)lk * NNEUR + (Nb + lr);
#pragma unroll 2
        for (int k = 0; k < INDIM; k += 4) {
            v2f fa0 = *(const v2f*)(a0 + k);
            v2f fa1 = *(const v2f*)(a1 + k);
            v2f fb[4];
#pragma unroll
            for (int j = 0; j < 4; ++j) {
                const float* q = bp + (size_t)k * NNEUR + j * 16;
                fb[j] = v2f{q[0], q[NNEUR]};
            }
#pragma unroll
            for (int j = 0; j < 4; ++j) {
                c[0][j] = WMMA_F32(fa0, fb[j], c[0][j]);
                c[1][j] = WMMA_F32(fa1, fb[j], c[1][j]);
            }
        }
    }

    // ---- rvs @ U^T : K = 64.  B[j,n] = U[n*64+j] -> k-major per lane ----
    {
        const float* a0 = rvs + (size_t)(Mb + lr)      * RANKD + lk;
        const float* a1 = rvs + (size_t)(Mb + 16 + lr) * RANKD + lk;
#pragma unroll 2
        for (int k = 0; k < RANKD; k += 4) {
            v2f fa0 = *(const v2f*)(a0 + k);
            v2f fa1 = *(const v2f*)(a1 + k);
            v2f fb[4];
#pragma unroll
            for (int j = 0; j < 4; ++j)
                fb[j] = *(const v2f*)(U + (size_t)(Nb + j * 16 + lr) * RANKD + lk + k);
#pragma unroll
            for (int j = 0; j < 4; ++j) {
                c[0][j] = WMMA_F32(fa0, fb[j], c[0][j]);
                c[1][j] = WMMA_F32(fa1, fb[j], c[1][j]);
            }
        }
    }

    // ---- Epilogue.  C/D layout: VGPR e -> M = base + e + 8*(lane>=16),
    //                             N = base + (lane&15)  -------------------
    const size_t half = (size_t)BATCHD * NNEUR;
#pragma unroll
    for (int i = 0; i < 2; ++i) {
        const int mbase = Mb + i * 16 + ((lane >> 4) << 3);
#pragma unroll
        for (int j = 0; j < 4; ++j) {
            const int n  = Nb + j * 16 + lr;
            const float Bv = Bb[n];
#pragma unroll
            for (int e = 0; e < 8; ++e) {
                const size_t idx = (size_t)(mbase + e) * NNEUR + n;
                const float xv    = x[idx];
                const float pre   = c[i][j][e] - xv + Bv;    // -x + mm + B (+ext +rec in c)
                const float noisy = pre + SIGMA_C * eps[idx];
                const float xn    = xv + DT_TAU * noisy;
                out[idx] = xn;
                // numerically stable softplus: max(z,0) + log1p(exp(-|z|))
                const float sp = fmaxf(xn, 0.0f) + log1pf(expf(-fabsf(xn)));
                out[half + idx] = sp;
            }
        }
    }
}

// ---------------------------------------------------------------------------
// Launch
// ---------------------------------------------------------------------------
extern "C" void kernel_launch(void* const* d_in, const int* in_sizes, int n_in,
                              void* d_out, int out_size, void* d_ws, size_t ws_size,
                              hipStream_t stream) {
    // setup_inputs() order:
    const float* r_th  = (const float*)d_in[0];   // (2048, 64)
    const float* stim  = (const float*)d_in[1];   // (2048, 128)
    const float* x     = (const float*)d_in[2];   // (2048, 2048)
    const float* r     = (const float*)d_in[3];   // (2048, 2048)
    const float* J     = (const float*)d_in[4];   // (2048, 2048)
    const float* Bb    = (const float*)d_in[5];   // (1, 2048)
    const float* U     = (const float*)d_in[6];   // (2048, 64)
    const float* V     = (const float*)d_in[7];   // (64, 2048)
    const float* Ist   = (const float*)d_in[8];   // (128, 2048)
    const float* eps   = (const float*)d_in[9];   // (2048, 2048)
    float*       out   = (float*)d_out;
    float*       rvs   = (float*)d_ws;            // 2048*64*4 = 512 KB scratch

    // Kernel 1: 512 tiles (128 M-tiles x 4 rank-tiles), 8 waves/block
    rnn_rv_kernel<<<64, 256, 0, stream>>>(r, V, r_th, rvs);

    // Kernel 2: 2048 wave-tiles of 32x64, 8 waves/block
    rnn_main_kernel<<<256, 256, 0, stream>>>(r, J, stim, Ist, rvs, U,
                                             x, Bb, eps, out);
}